// PChooseLayer_34626026340799
// MI455X (gfx1250) — compile-verified
//
#include <hip/hip_runtime.h>
#include <hip/hip_bf16.h>

// B=8, TGT=256, SRC=512, D=1024, H=16, dh=64, TEMP=0.2
// d_out = [p_choose (8*16*256*512) | alpha (same)]  (float32)
// d_ws  = [Q (2048*1024 f32) | K (4096*1024 f32)]  (24 MB)

typedef __attribute__((ext_vector_type(2))) float v2f;
typedef __attribute__((ext_vector_type(8))) float v8f;

#define P_ELEMS (8 * 16 * 256 * 512)

// ---------------------------------------------------------------------------
// Kernel 1: C[m,n] = relu( sum_d A[m,d]*W[n,d] + bias[n] ),  N = D = 1024.
// One wave per 16x16 tile, f32 WMMA 16x16x4 chained over K.
// A frag: lane (h*16+m) holds A[m, k0+2h], A[m, k0+2h+1]  (VGPR0, VGPR1)
// B frag: lane (h*16+n) holds W[n, k0+2h], W[n, k0+2h+1]
// ---------------------------------------------------------------------------
__global__ __launch_bounds__(256) void proj_relu_kernel(
    const float* __restrict__ A, const float* __restrict__ W,
    const float* __restrict__ bias, float* __restrict__ C)
{
    const int lane = threadIdx.x & 31;
    const int wave = threadIdx.x >> 5;
    const int tile = blockIdx.x * 8 + wave;   // (M/16)*64 tiles total
    const int mt = tile >> 6;                 // N-tiles = 1024/16 = 64
    const int nt = tile & 63;
    const int hi = lane >> 4;                 // half-wave -> K pair select
    const int lm = lane & 15;                 // M (for A) / N (for B) row

    const float* arow = A + (size_t)(mt * 16 + lm) * 1024 + 2 * hi;
    const float* wrow = W + (size_t)(nt * 16 + lm) * 1024 + 2 * hi;

    v8f acc = {};
    for (int k0 = 0; k0 < 1024; k0 += 4) {
        v2f av = *(const v2f*)(arow + k0);
        v2f bv = *(const v2f*)(wrow + k0);
        acc = __builtin_amdgcn_wmma_f32_16x16x4_f32(
            false, av, false, bv, (short)0, acc, false, false);
    }

    const int col = nt * 16 + lm;
    const float bb = bias[col];
    float* crow = C + (size_t)(mt * 16 + hi * 8) * 1024 + col;
#pragma unroll
    for (int j = 0; j < 8; ++j) {             // VGPR j holds M = hi*8 + j
        float v = acc[j] + bb;
        crow[(size_t)j * 1024] = v > 0.0f ? v : 0.0f;
    }
}

// ---------------------------------------------------------------------------
// Kernel 2: per (b,h): energy = Q_h * K_h^T * (1/8) + eb;  p = sigmoid(5*e)
// One wave per 16x16 tile of the 256x512 energy matrix, K = dh = 64.
// ---------------------------------------------------------------------------
__global__ __launch_bounds__(256) void energy_p_kernel(
    const float* __restrict__ Q, const float* __restrict__ K,
    const float* __restrict__ eb, float* __restrict__ P)
{
    const int lane = threadIdx.x & 31;
    const int wave = threadIdx.x >> 5;
    const int tile = blockIdx.x * 8 + wave;   // 128 * 16 * 32 = 65536 tiles
    const int bh  = tile >> 9;
    const int rem = tile & 511;
    const int tt  = rem >> 5;                 // 16 target tiles
    const int st  = rem & 31;                 // 32 source tiles
    const int b = bh >> 4;
    const int h = bh & 15;
    const int hi = lane >> 4;
    const int lm = lane & 15;

    const float* qrow = Q + (size_t)(b * 256 + tt * 16 + lm) * 1024 + h * 64 + 2 * hi;
    const float* krow = K + (size_t)(b * 512 + st * 16 + lm) * 1024 + h * 64 + 2 * hi;

    v8f acc = {};
#pragma unroll
    for (int k0 = 0; k0 < 64; k0 += 4) {
        v2f av = *(const v2f*)(qrow + k0);
        v2f bv = *(const v2f*)(krow + k0);
        acc = __builtin_amdgcn_wmma_f32_16x16x4_f32(
            false, av, false, bv, (short)0, acc, false, false);
    }

    const float bias = eb[0];
    const int col = st * 16 + lm;
    float* prow = P + ((size_t)bh * 256 + tt * 16 + hi * 8) * 512 + col;
#pragma unroll
    for (int j = 0; j < 8; ++j) {
        float e = acc[j] * 0.125f + bias;     // dh^-0.5 = 1/8
        float x = e * 5.0f;                   // 1/TEMPERATURE
        prow[(size_t)j * 512] = 1.0f / (1.0f + __expf(-x));
    }
}

// ---------------------------------------------------------------------------
// Kernel 3: monotonic alignment. One block per (b,h). 128 threads, 4 src
// elems each. Per target step t (sequential, 256 steps):
//   q_s = a_s * q_{s-1} + b_s,  a_s = (s==0 ? 0 : 1 - p_t[s-1]),
//   b_s = alpha_{t-1}[s]  (alpha_{-1} = one-hot at s=0),  alpha_t[s]=p_t[s]*q_s
// Solved as a parallel scan over (a,b) pairs: 4 serial per lane, 5 shfl_up
// steps intra-wave, 4-wave LDS combine. q_{-1}=0 => q = B of composed prefix.
// ---------------------------------------------------------------------------
__global__ __launch_bounds__(128) void align_kernel(
    const float* __restrict__ P, float* __restrict__ Alpha)
{
    const int bh   = blockIdx.x;
    const int tid  = threadIdx.x;        // 0..127
    const int lane = tid & 31;
    const int wv   = tid >> 5;
    const int s0   = tid * 4;

    __shared__ float sA[4];
    __shared__ float sB[4];

    float aprev[4];
#pragma unroll
    for (int e = 0; e < 4; ++e) aprev[e] = (s0 + e == 0) ? 1.0f : 0.0f;

    const float* pb = P + (size_t)bh * (256 * 512);
    float* ab = Alpha + (size_t)bh * (256 * 512);

    for (int t = 0; t < 256; ++t) {
        const float* prow = pb + t * 512;
        float4 pv = *(const float4*)(prow + s0);
        float pvals[4] = {pv.x, pv.y, pv.z, pv.w};
        float pm1 = 0.0f;
        if (s0 != 0) pm1 = prow[s0 - 1];   // predicated (no OOB at s0==0)

        // thread-local prefix composition (A,B) per element
        float pa[4], pbv[4];
        float Ac = 1.0f, Bc = 0.0f;
#pragma unroll
        for (int e = 0; e < 4; ++e) {
            float prevp = (e == 0) ? pm1 : pvals[e - 1];
            float a = (s0 + e == 0) ? 0.0f : (1.0f - prevp);
            float b = aprev[e];
            Bc = a * Bc + b;
            Ac = a * Ac;
            pa[e] = Ac;
            pbv[e] = Bc;
        }

        // intra-wave inclusive scan of thread aggregates (Ac,Bc)
        float sa = Ac, sb = Bc;
#pragma unroll
        for (int d = 1; d < 32; d <<= 1) {
            float ua = __shfl_up(sa, d, 32);
            float ub = __shfl_up(sb, d, 32);
            if (lane >= d) { sb = sa * ub + sb; sa = sa * ua; }
        }
        // thread-exclusive within wave
        float ea = __shfl_up(sa, 1, 32);
        float eb = __shfl_up(sb, 1, 32);
        if (lane == 0) { ea = 1.0f; eb = 0.0f; }

        // per-wave aggregate -> LDS
        if (lane == 31) { sA[wv] = sa; sB[wv] = sb; }
        __syncthreads();

        // exclusive composition of earlier waves
        float wa = 1.0f, wb = 0.0f;
        for (int w = 0; w < wv; ++w) { wb = sA[w] * wb + sB[w]; wa = sA[w] * wa; }
        __syncthreads();   // LDS reusable next t

        // full thread-exclusive prefix: (ea,eb) after (wa,wb)
        float xb = ea * wb + eb;   // q entering this thread (q_{-1}=0)

        float4 av;
        float out[4];
#pragma unroll
        for (int e = 0; e < 4; ++e) {
            float q = pa[e] * xb + pbv[e];
            out[e] = pvals[e] * q;
            aprev[e] = out[e];
        }
        av.x = out[0]; av.y = out[1]; av.z = out[2]; av.w = out[3];
        *(float4*)(ab + t * 512 + s0) = av;
    }
}

extern "C" void kernel_launch(void* const* d_in, const int* in_sizes, int n_in,
                              void* d_out, int out_size, void* d_ws, size_t ws_size,
                              hipStream_t stream) {
    const float* seqs = (const float*)d_in[0];
    const float* keys = (const float*)d_in[1];
    const float* q_w  = (const float*)d_in[2];
    const float* q_b  = (const float*)d_in[3];
    const float* k_w  = (const float*)d_in[4];
    const float* k_b  = (const float*)d_in[5];
    const float* e_b  = (const float*)d_in[6];

    float* out   = (float*)d_out;
    float* P     = out;              // p_choose
    float* Alpha = out + P_ELEMS;    // alpha

    float* Qbuf = (float*)d_ws;              // 2048*1024 f32
    float* Kbuf = Qbuf + 2048 * 1024;        // 4096*1024 f32

    // Q = relu(seqs @ q_w^T + q_b): 2048x1024 -> 128*64/8 = 1024 blocks
    proj_relu_kernel<<<1024, 256, 0, stream>>>(seqs, q_w, q_b, Qbuf);
    // K = relu(keys @ k_w^T + k_b): 4096x1024 -> 256*64/8 = 2048 blocks
    proj_relu_kernel<<<2048, 256, 0, stream>>>(keys, k_w, k_b, Kbuf);
    // energy + sigmoid: 65536 tiles / 8 waves = 8192 blocks
    energy_p_kernel<<<8192, 256, 0, stream>>>(Qbuf, Kbuf, e_b, P);
    // monotonic alignment: one block per (b,h)
    align_kernel<<<128, 128, 0, stream>>>(P, Alpha);
}